// EfficientBlockCirculantLinear_34368328303059
// MI455X (gfx1250) — compile-verified
//
#include <hip/hip_runtime.h>
#include <math.h>

// ---------------------------------------------------------------------------
// Block-circulant linear layer, frequency-domain, all-WMMA (gfx1250, wave32)
//   BATCH=1024, D_IN=D_OUT=4096, B=128, K_IN=K_OUT=32, bins 0..64 (rfft)
//   y = iDFT( (DFT(x.*D per 128-block) @ conj(DFT(W))) )   -- three GEMM stages
// ---------------------------------------------------------------------------

#define TWO_PI 6.28318530717958647692f

typedef float v2f __attribute__((ext_vector_type(2)));
typedef float v8f __attribute__((ext_vector_type(8)));

// D = A(16x4) * B(4x16) + C(16x16), fp32 end-to-end.
__device__ __forceinline__ v8f wmma4(v2f a, v2f b, v8f c) {
  return __builtin_amdgcn_wmma_f32_16x16x4_f32(
      /*neg_a=*/false, a, /*neg_b=*/false, b,
      /*c_mod=*/(short)0, c, /*reuse_a=*/false, /*reuse_b=*/false);
}

// ---------------------------------------------------------------------------
// Kernel 1a: conj(FFT(W)) -> Wr/Wi laid out as GEMM B-matrices [bin][j][o].
// ---------------------------------------------------------------------------
__global__ void bc_wfft_kernel(const float* __restrict__ W,
                               float* __restrict__ Wr, float* __restrict__ Wi) {
  int idx = blockIdx.x * blockDim.x + threadIdx.x;
  if (idx >= 65 * 32 * 32) return;
  int o = idx & 31;
  int j = (idx >> 5) & 31;
  int bin = idx >> 10;
  const float* w = W + (o * 32 + j) * 128;
  float cr = 0.f, ci = 0.f;
  for (int u = 0; u < 128; ++u) {
    int ph = (bin * u) & 127;  // exact phase reduction
    float s, c;
    sincosf((float)ph * (TWO_PI / 128.f), &s, &c);
    cr = fmaf(w[u], c, cr);
    ci = fmaf(w[u], s, ci);
  }
  Wr[(bin * 32 + j) * 32 + o] = cr;
  Wi[(bin * 32 + j) * 32 + o] = ci;
}

// ---------------------------------------------------------------------------
// Kernel 1b: DFT tables F (forward, 128x144) and G (inverse, 132x128).
// ---------------------------------------------------------------------------
__global__ void bc_tables_kernel(float* __restrict__ F, float* __restrict__ G) {
  int idx = blockIdx.x * blockDim.x + threadIdx.x;
  if (idx < 128 * 144) {
    int col = idx % 144, u = idx / 144;
    float val = 0.f, s, c;
    if (col < 65) {
      int ph = (u * col) & 127;
      sincosf((float)ph * (TWO_PI / 128.f), &s, &c);
      val = c;
    } else if (col >= 72 && col < 137) {
      int b = col - 72;
      int ph = (u * b) & 127;
      sincosf((float)ph * (TWO_PI / 128.f), &s, &c);
      val = -s;
    }
    F[idx] = val;
    return;
  }
  int gidx = idx - 128 * 144;
  if (gidx < 132 * 128) {
    int t = gidx & 127, k = gidx >> 7;
    float val = 0.f;
    if (k < 130) {
      int b = k >> 1, ri = k & 1;
      float cb = (b == 0 || b == 64) ? 1.f : 2.f;
      int ph = (b * t) & 127;
      float s, c;
      sincosf((float)ph * (TWO_PI / 128.f), &s, &c);
      val = (ri == 0 ? cb * c : -cb * s) * (1.f / 128.f);
    }
    G[gidx] = val;
  }
}

// ---------------------------------------------------------------------------
// Kernel 2: Xf = (x .* D) blocks @ F.
//   Preload fragments, then two independent WMMA chains (even/odd K-steps)
//   so the scheduler can keep multiple LDS loads in flight.
// ---------------------------------------------------------------------------
__global__ __launch_bounds__(128) void bc_dft_kernel(
    const float* __restrict__ x, const float* __restrict__ D,
    const float* __restrict__ F, float* __restrict__ Xf) {
  extern __shared__ float smem[];
  float* sx = smem;             // 64 x 128
  float* sF = smem + 64 * 128;  // 128 x 144
  const int j = blockIdx.y;
  const int nb = blockIdx.x * 64;
  const int tid = threadIdx.x;

  for (int i = tid; i < 64 * 128; i += 128) {
    int r = i >> 7, c = i & 127;
    sx[i] = x[(nb + r) * 4096 + j * 128 + c] * D[j * 128 + c];
  }
  for (int i = tid; i < 128 * 144; i += 128) sF[i] = F[i];
  __syncthreads();

  const int wave = tid >> 5, lane = tid & 31;
  const int lo = lane & 15, hi = lane >> 4;
  const int m0 = wave * 16;  // 16 batch rows per wave

  // A fragments are invariant across the 9 N-tiles: load once.
  v2f af[32];
#pragma unroll
  for (int ks = 0; ks < 32; ++ks) {
    const int ka = ks * 4 + 2 * hi;
    const float* p = &sx[(m0 + lo) * 128 + ka];
    af[ks].x = p[0];
    af[ks].y = p[1];
  }

  for (int nt = 0; nt < 9; ++nt) {
    v2f bf[32];
#pragma unroll
    for (int ks = 0; ks < 32; ++ks) {
      const int ka = ks * 4 + 2 * hi;
      bf[ks].x = sF[ka * 144 + nt * 16 + lo];
      bf[ks].y = sF[(ka + 1) * 144 + nt * 16 + lo];
    }
    v8f acc0 = {}, acc1 = {};
#pragma unroll
    for (int ks = 0; ks < 32; ks += 2) {  // two independent chains
      acc0 = wmma4(af[ks], bf[ks], acc0);
      acc1 = wmma4(af[ks + 1], bf[ks + 1], acc1);
    }
    v8f acc = acc0 + acc1;  // identical D layouts -> exact elementwise merge

    const int n = nt * 16 + lo;
    int ri = 0, bin = n;
    bool valid = true;
    if (n >= 72 && n < 137) { ri = 1; bin = n - 72; }
    else if (n >= 65) valid = false;
    if (valid) {
#pragma unroll
      for (int v = 0; v < 8; ++v) {
        int m = v + 8 * hi;
        Xf[((bin * 32 + j) * 2 + ri) * 1024 + nb + m0 + m] = acc[v];
      }
    }
  }
}

// ---------------------------------------------------------------------------
// Kernel 3: per-bin complex einsum (K=32) + iDFT GEMM (K=132), fused.
//   Stage B: batched global preload -> 4 independent WMMA chains.
//   Stage C: G fragments hoisted per t-tile, 2 independent chains.
// ---------------------------------------------------------------------------
__global__ __launch_bounds__(128) void bc_einsum_idft_kernel(
    const float* __restrict__ Xf, const float* __restrict__ Wr,
    const float* __restrict__ Wi, const float* __restrict__ G,
    float* __restrict__ out) {
  extern __shared__ float smem[];
  float* sYf = smem;  // 16 * 132 * 17 floats
  const int nb = blockIdx.x * 16;
  const int ob = blockIdx.y * 16;
  const int tid = threadIdx.x;
  const int wave = tid >> 5, lane = tid & 31;
  const int lo = lane & 15, hi = lane >> 4;

  // Zero the K-padding slices (k = 130, 131) so 0*garbage never makes NaN.
  for (int i = tid; i < 16 * 2 * 17; i += 128) {
    int o = i / 34, rem = i % 34;
    int k = 130 + rem / 17, m = rem % 17;
    sYf[(o * 132 + k) * 17 + m] = 0.f;
  }
  __syncthreads();

  // ---- Stage B: Yf[b] = Xf[b] @ conj(Wfft)[b], complex, K=32, per bin ----
  for (int bin = wave; bin < 65; bin += 4) {
    v2f Ar[8], Ai[8], Br[8], Bs[8];
#pragma unroll
    for (int ks = 0; ks < 8; ++ks) {
      const int ka = ks * 4 + 2 * hi;
      const float* xr = Xf + ((bin * 32 + ka) * 2 + 0) * 1024 + nb + lo;
      const float* xi = Xf + ((bin * 32 + ka) * 2 + 1) * 1024 + nb + lo;
      Ar[ks].x = xr[0];    Ar[ks].y = xr[2048];   // next j: +2*1024 floats
      Ai[ks].x = xi[0];    Ai[ks].y = xi[2048];
      const float* wr = Wr + (bin * 32 + ka) * 32 + ob + lo;
      const float* wsv = Wi + (bin * 32 + ka) * 32 + ob + lo;
      Br[ks].x = wr[0];    Br[ks].y = wr[32];     // next j: +32 floats
      Bs[ks].x = wsv[0];   Bs[ks].y = wsv[32];
    }
    v8f accR0 = {}, accR1 = {}, accI0 = {}, accI1 = {};
#pragma unroll
    for (int ks = 0; ks < 8; ks += 2) {  // four independent chains
      v2f AiN0 = -Ai[ks];      // f32 WMMA has no A/B NEG -> VALU negate
      v2f AiN1 = -Ai[ks + 1];
      accR0 = wmma4(Ar[ks],     Br[ks],     accR0);
      accI0 = wmma4(Ar[ks],     Bs[ks],     accI0);
      accR1 = wmma4(Ar[ks + 1], Br[ks + 1], accR1);
      accI1 = wmma4(Ar[ks + 1], Bs[ks + 1], accI1);
      accR0 = wmma4(AiN0,       Bs[ks],     accR0);
      accI0 = wmma4(Ai[ks],     Br[ks],     accI0);
      accR1 = wmma4(AiN1,       Bs[ks + 1], accR1);
      accI1 = wmma4(Ai[ks + 1], Br[ks + 1], accI1);
    }
    v8f accR = accR0 + accR1;
    v8f accI = accI0 + accI1;
#pragma unroll
    for (int v = 0; v < 8; ++v) {
      int m = v + 8 * hi;  // batch row within tile
      sYf[(lo * 132 + 2 * bin + 0) * 17 + m] = accR[v];
      sYf[(lo * 132 + 2 * bin + 1) * 17 + m] = accI[v];
    }
  }
  __syncthreads();

  // ---- Stage C: y[n, o, :] = Yf_o[n, k] @ G[k, t], K=132, N=128 ----
  for (int tt = 0; tt < 8; ++tt) {
    v2f bg[33];  // G fragments: shared by all o-columns this wave handles
#pragma unroll
    for (int ks = 0; ks < 33; ++ks) {
      const int ka = ks * 4 + 2 * hi;
      bg[ks].x = G[ka * 128 + tt * 16 + lo];
      bg[ks].y = G[(ka + 1) * 128 + tt * 16 + lo];
    }
    for (int oi = wave; oi < 16; oi += 4) {
      v2f ay[33];
#pragma unroll
      for (int ks = 0; ks < 33; ++ks) {
        const int ka = ks * 4 + 2 * hi;
        ay[ks].x = sYf[(oi * 132 + ka) * 17 + lo];
        ay[ks].y = sYf[(oi * 132 + ka + 1) * 17 + lo];
      }
      v8f acc0 = {}, acc1 = {};
#pragma unroll
      for (int ks = 0; ks < 32; ks += 2) {  // two independent chains
        acc0 = wmma4(ay[ks], bg[ks], acc0);
        acc1 = wmma4(ay[ks + 1], bg[ks + 1], acc1);
      }
      acc0 = wmma4(ay[32], bg[32], acc0);   // K-step 33 (zero-padded tail)
      v8f acc = acc0 + acc1;
#pragma unroll
      for (int v = 0; v < 8; ++v) {
        int m = v + 8 * hi;
        out[(nb + m) * 4096 + (ob + oi) * 128 + tt * 16 + lo] = acc[v];
      }
    }
  }
}

// ---------------------------------------------------------------------------
// Host-side launcher. Workspace (floats):
//   Wr 66560 | Wi 66560 | F 18432 | G 16896 | Xf 65*32*2*1024 = 4,259,840
//   total = 4,428,288 floats (~17.7 MB)
// ---------------------------------------------------------------------------
extern "C" void kernel_launch(void* const* d_in, const int* in_sizes, int n_in,
                              void* d_out, int out_size, void* d_ws, size_t ws_size,
                              hipStream_t stream) {
  (void)in_sizes; (void)n_in; (void)out_size; (void)ws_size;
  const float* x = (const float*)d_in[0];
  const float* W = (const float*)d_in[1];
  const float* D = (const float*)d_in[2];
  float* out = (float*)d_out;

  float* ws = (float*)d_ws;
  float* Wr = ws;
  float* Wi = Wr + 65 * 32 * 32;
  float* F  = Wi + 65 * 32 * 32;
  float* G  = F + 128 * 144;
  float* Xf = G + 132 * 128;

  bc_wfft_kernel<<<(65 * 32 * 32 + 255) / 256, 256, 0, stream>>>(W, Wr, Wi);
  bc_tables_kernel<<<(128 * 144 + 132 * 128 + 255) / 256, 256, 0, stream>>>(F, G);

  dim3 g2(16, 32);
  size_t lds2 = (size_t)(64 * 128 + 128 * 144) * sizeof(float);  // ~106.5 KB
  bc_dft_kernel<<<g2, 128, lds2, stream>>>(x, D, F, Xf);

  dim3 g3(64, 2);
  size_t lds3 = (size_t)(16 * 132 * 17) * sizeof(float);         // ~143.6 KB
  bc_einsum_idft_kernel<<<g3, 128, lds3, stream>>>(Xf, Wr, Wi, G, out);
}